// MATLABStyleEnhancement_88819923681601
// MI455X (gfx1250) — compile-verified
//
#include <hip/hip_runtime.h>

// ---------------------------------------------------------------------------
// MATLAB-style dehaze/enhance for MI455X (gfx1250, wave32).
//  - Box filters (reflect pad, radius r) done as banded GEMMs with
//    V_WMMA_F32_16X16X4_F32: per 16x16 output tile only the K-band
//    (width 2r+16 -> ceil/4 WMMAs) is contracted.
//  - Percentiles via exact 4x8-bit radix-select on float bit patterns
//    (J in [0,1] => bit order == numeric order), no sort.
//  - Everything L2-resident (192MB L2); ~112MB workspace, reused planes.
// ---------------------------------------------------------------------------

constexpr int IMW   = 512;
constexpr int IMH   = 512;
constexpr int NB    = 8;
constexpr int HWPX  = IMW * IMH;          // 262144 pixels per image
constexpr int PLANE = NB * HWPX;          // one (B,1,H,W) plane: 2,097,152 floats

typedef float v2f __attribute__((ext_vector_type(2)));
typedef float v8f __attribute__((ext_vector_type(8)));

__device__ __forceinline__ float clampf(float x, float lo, float hi) {
  return fminf(fmaxf(x, lo), hi);
}
__device__ __forceinline__ int clampi(int x, int lo, int hi) {
  return x < lo ? lo : (x > hi ? hi : x);
}

// ---------------------------------------------------------------------------
// Build the reflect-pad box matrices: M[i,j] = #{d in [-r,r] : reflect(j+d)==i}
// Horizontal pass: Out = X * M.  Vertical pass: Out = M^T * X  (Mt stored).
// ---------------------------------------------------------------------------
__global__ void k_build_M(float* __restrict__ Mm, float* __restrict__ Mt,
                          const int* __restrict__ rptr) {
  int idx = blockIdx.x * blockDim.x + threadIdx.x;
  if (idx >= HWPX) return;
  int i = idx >> 9, j = idx & 511;
  int r = clampi(rptr[0], 1, 50);
  float cnt = 0.f;
  for (int d = -r; d <= r; ++d) {
    int t = j + d;                       // jnp.pad mode='reflect' (no edge dup)
    if (t < 0) t = -t;
    else if (t > IMW - 1) t = 2 * (IMW - 1) - t;
    cnt += (t == i) ? 1.f : 0.f;
  }
  Mm[i * IMW + j] = cnt;                 // M[input i][output j]
  Mt[j * IMW + i] = cnt;                 // Mt[output j][input i]
}

// ---------------------------------------------------------------------------
// dark channel -> t0, plus grayscale guide
// ---------------------------------------------------------------------------
__global__ void k_dark_gray(const float* __restrict__ img, const float* __restrict__ omega,
                            const float* __restrict__ atm, float* __restrict__ gray,
                            float* __restrict__ t0) {
  int idx = blockIdx.x * blockDim.x + threadIdx.x;
  if (idx >= PLANE) return;
  int b = idx / HWPX, pix = idx - b * HWPX;
  const float* ib = img + (size_t)b * 3 * HWPX + pix;
  float r0 = ib[0], g0 = ib[HWPX], b0 = ib[2 * HWPX];
  float a0 = atm[b * 3 + 0], a1 = atm[b * 3 + 1], a2 = atm[b * 3 + 2];
  float dark = fminf(fminf(r0 / (a0 + 1e-8f), g0 / (a1 + 1e-8f)), b0 / (a2 + 1e-8f));
  t0[idx]   = 1.f - omega[b] * dark;
  gray[idx] = 0.299f * r0 + 0.587f * g0 + 0.114f * b0;
}

// ---------------------------------------------------------------------------
// gradient-constrained tg (pure neighbor reads, no scan dependency), plus
// the box-filter input products gray*tg and gray*gray
// ---------------------------------------------------------------------------
__global__ void k_tg(const float* __restrict__ t0, const float* __restrict__ gray,
                     float* __restrict__ tg, float* __restrict__ f2, float* __restrict__ f3) {
  int idx = blockIdx.x * blockDim.x + threadIdx.x;
  if (idx >= PLANE) return;
  int b = idx / HWPX, pix = idx - b * HWPX;
  int y = pix >> 9, x = pix & 511;
  const float* tb = t0 + (size_t)b * HWPX;
  float cur = tb[y * IMW + x];
  float out;
  if (y == 0) {
    if (x == 0) out = cur;
    else { float l = tb[x - 1]; out = l * expf(-fabsf(cur - l)); }
  } else {
    float up = tb[(y - 1) * IMW + x];
    float tx_up;
    if (x == 0) tx_up = up;
    else { float ul = tb[(y - 1) * IMW + x - 1]; tx_up = ul * expf(-fabsf(up - ul)); }
    out = tx_up * expf(-fabsf(cur - up));
  }
  tg[idx] = out;
  float g = gray[idx];
  f2[idx] = g * out;
  f3[idx] = g * g;
}

// ---------------------------------------------------------------------------
// Banded box-filter GEMM pass using V_WMMA_F32_16X16X4_F32.
//  vertical==0: Out(z) = X(z) * Mm      (band centered on output column tile)
//  vertical==1: Out(z) = Mt   * X(z)    (band centered on output row tile,
//                                        result scaled by 1/(2r+1)^2)
// One wave (32 threads) per 16x16 output tile; K-band = 4*KT >= 2r+16.
// A 16x4 frag: lane m<16 -> A[m, k..k+1], lanes 16..31 -> A[m, k+2..k+3].
// C/D: VGPR v -> rows v (lanes 0-15) and v+8 (lanes 16-31).
// ---------------------------------------------------------------------------
__global__ void k_box_wmma(const float* __restrict__ imgMat, const float* __restrict__ sharedMat,
                           float* __restrict__ Out, const int* __restrict__ rptr, int vertical) {
  int j0 = blockIdx.x * 16;
  int r0 = blockIdx.y * 16;
  int z  = blockIdx.z;
  int r  = clampi(rptr[0], 1, 50);
  int KT = (2 * r + 19) >> 2;                     // ceil((2r+16)/4)
  int center = vertical ? r0 : j0;
  int c0 = center - r;
  if (c0 < 0) c0 = 0;
  int cmax = IMW - 4 * KT;
  if (c0 > cmax) c0 = cmax;

  int off = z * HWPX;
  const float* Ap;
  const float* Bp;
  if (vertical) { Ap = sharedMat; Bp = imgMat + off; }
  else          { Ap = imgMat + off; Bp = sharedMat; }

  int lane = threadIdx.x & 31;
  int lm = lane & 15;
  int lh = lane >> 4;

  v8f acc = {0.f, 0.f, 0.f, 0.f, 0.f, 0.f, 0.f, 0.f};
  for (int t = 0; t < KT; ++t) {
    int k = c0 + 4 * t + 2 * lh;
    v2f a, bv;
    a.x  = Ap[(r0 + lm) * IMW + k];
    a.y  = Ap[(r0 + lm) * IMW + k + 1];
    bv.x = Bp[k * IMW + j0 + lm];
    bv.y = Bp[(k + 1) * IMW + j0 + lm];
    acc = __builtin_amdgcn_wmma_f32_16x16x4_f32(false, a, false, bv,
                                                (short)0, acc, false, false);
  }
  float kk = (float)((2 * r + 1) * (2 * r + 1));
  float scale = vertical ? (1.f / kk) : 1.f;
  float* o = Out + off;
#pragma unroll
  for (int v = 0; v < 8; ++v)
    o[(r0 + v + 8 * lh) * IMW + j0 + lm] = acc[v] * scale;
}

// ---------------------------------------------------------------------------
// guided-filter coefficients a, b (eps = 0.5)
// ---------------------------------------------------------------------------
__global__ void k_ab(const float* __restrict__ mI, const float* __restrict__ mp,
                     const float* __restrict__ mIp, const float* __restrict__ mII,
                     float* __restrict__ a, float* __restrict__ b) {
  int idx = blockIdx.x * blockDim.x + threadIdx.x;
  if (idx >= PLANE) return;
  float mi = mI[idx], mpv = mp[idx];
  float cov = mIp[idx] - mi * mpv;
  float var = mII[idx] - mi * mi;
  float av = cov / (var + 0.5f);
  a[idx] = av;
  b[idx] = mpv - av * mi;
}

// ---------------------------------------------------------------------------
// t_final (kept in registers) + restoration -> J written straight to d_out
// ---------------------------------------------------------------------------
__global__ void k_restore(const float* __restrict__ img, const float* __restrict__ atm,
                          const float* __restrict__ gray, const float* __restrict__ boxA,
                          const float* __restrict__ boxB, float* __restrict__ J) {
  int idx = blockIdx.x * blockDim.x + threadIdx.x;
  if (idx >= PLANE) return;
  int b = idx / HWPX, pix = idx - b * HWPX;
  float tf = clampf(boxA[idx] * gray[idx] + boxB[idx], 0.1f, 1.0f);
  float den = tf + 1e-8f;
#pragma unroll
  for (int c = 0; c < 3; ++c) {
    float Ac = atm[b * 3 + c];
    size_t o = (size_t)(b * 3 + c) * HWPX + pix;
    J[o] = clampf((img[o] - Ac) / den + Ac, 0.f, 1.f);
  }
}

// ---------------------------------------------------------------------------
// Exact k-th order statistic per (b,c) channel via 4-pass 8-bit radix select
// on float bit patterns (all values >= 0). One block per (channel, lo/hi).
// ---------------------------------------------------------------------------
__global__ void k_select(const float* __restrict__ J, const float* __restrict__ Llow,
                         const float* __restrict__ Lhigh, float* __restrict__ P) {
  __shared__ unsigned int hist[256];
  __shared__ unsigned int sPrefix;
  __shared__ int sK;
  const int n = HWPX;
  int ch = blockIdx.x >> 1;
  int which = blockIdx.x & 1;
  int b = ch / 3;
  int tid = threadIdx.x;
  if (tid == 0) {
    float Lv = which ? Lhigh[b] : Llow[b];
    sK = clampi((int)(Lv / 100.0f * (float)n), 0, n - 1);  // trunc like astype(int32)
    sPrefix = 0u;
  }
  const float* base = J + (size_t)ch * n;
  for (int s = 24; s >= 0; s -= 8) {
    hist[tid] = 0u;
    __syncthreads();
    unsigned int pref = sPrefix;
    unsigned int maskHi = (s == 24) ? 0u : (0xFFFFFFFFu << (s + 8));
    for (int i = tid; i < n; i += 256) {
      unsigned int key = __float_as_uint(base[i]);
      if ((key & maskHi) == pref) atomicAdd(&hist[(key >> s) & 255u], 1u);
    }
    __syncthreads();
    if (tid == 0) {
      int k = sK;
      unsigned int c = 0;
      int d = 0;
      for (; d < 256; ++d) {
        unsigned int nc = c + hist[d];
        if ((int)nc > k) break;
        c = nc;
      }
      sK = k - (int)c;
      sPrefix = pref | ((unsigned int)d << s);
    }
    __syncthreads();
  }
  if (tid == 0) P[ch * 2 + which] = __uint_as_float(sPrefix);
}

// ---------------------------------------------------------------------------
// in-place percentile stretch on d_out
// ---------------------------------------------------------------------------
__global__ void k_stretch(float* __restrict__ out, const float* __restrict__ P) {
  int idx = blockIdx.x * blockDim.x + threadIdx.x;
  if (idx >= 3 * PLANE) return;
  int ch = idx / HWPX;                   // b*3+c
  float plo = P[ch * 2], phi = P[ch * 2 + 1];
  float v = clampf(out[idx], plo, phi);
  out[idx] = clampf((v - plo) / (phi - plo + 1e-8f), 0.f, 1.f);
}

// ---------------------------------------------------------------------------
// Workspace layout (floats):
//   [0, 262144)          Mm
//   [262144, 524288)     Mt
//   [524288, 524544)     P (48 used) + pad
//   planes p0..p12, each PLANE floats:
//     p0 gray | p1 tg | p2 gray*tg | p3 gray^2 | p4 t0 -> a
//     p5 hx0 -> b | p6 hx1 -> Ha | p7 hx2 -> Hb | p8 hx3 -> boxA
//     p9 mI -> boxB | p10 mp | p11 mIp | p12 mII
// Total ~112 MB; everything rewritten each call (deterministic / poison-safe).
// ---------------------------------------------------------------------------
extern "C" void kernel_launch(void* const* d_in, const int* in_sizes, int n_in,
                              void* d_out, int out_size, void* d_ws, size_t ws_size,
                              hipStream_t stream) {
  (void)in_sizes; (void)n_in; (void)out_size; (void)ws_size;
  const float* img   = (const float*)d_in[0];
  const float* omega = (const float*)d_in[1];
  const float* atm   = (const float*)d_in[2];
  const float* Llow  = (const float*)d_in[3];
  const float* Lhigh = (const float*)d_in[4];
  const int*   rptr  = (const int*)d_in[5];
  float* J = (float*)d_out;

  float* ws = (float*)d_ws;
  float* Mm = ws;
  float* Mt = ws + HWPX;
  float* P  = ws + 2 * HWPX;
  float* pb = ws + 2 * HWPX + 256;
  auto plane = [&](int i) { return pb + (size_t)i * PLANE; };

  k_build_M<<<HWPX / 256, 256, 0, stream>>>(Mm, Mt, rptr);
  k_dark_gray<<<PLANE / 256, 256, 0, stream>>>(img, omega, atm, plane(0), plane(4));
  k_tg<<<PLANE / 256, 256, 0, stream>>>(plane(4), plane(0), plane(1), plane(2), plane(3));

  // Box round 1: fields {gray, tg, gray*tg, gray^2} batched as z = field*8 + b
  k_box_wmma<<<dim3(32, 32, 32), 32, 0, stream>>>(plane(0), Mm, plane(5), rptr, 0);
  k_box_wmma<<<dim3(32, 32, 32), 32, 0, stream>>>(plane(5), Mt, plane(9), rptr, 1);

  k_ab<<<PLANE / 256, 256, 0, stream>>>(plane(9), plane(10), plane(11), plane(12),
                                        plane(4), plane(5));

  // Box round 2: fields {a, b}
  k_box_wmma<<<dim3(32, 32, 16), 32, 0, stream>>>(plane(4), Mm, plane(6), rptr, 0);
  k_box_wmma<<<dim3(32, 32, 16), 32, 0, stream>>>(plane(6), Mt, plane(8), rptr, 1);

  k_restore<<<PLANE / 256, 256, 0, stream>>>(img, atm, plane(0), plane(8), plane(9), J);
  k_select<<<48, 256, 0, stream>>>(J, Llow, Lhigh, P);
  k_stretch<<<(3 * PLANE) / 256, 256, 0, stream>>>(J, P);
}